// NNarray_29987461661234
// MI455X (gfx1250) — compile-verified
//
#include <hip/hip_runtime.h>
#include <hip/hip_bf16.h>

typedef float v2f __attribute__((ext_vector_type(2)));
typedef float v8f __attribute__((ext_vector_type(8)));

#define CRXB_GMAX 0.000333f
#define CRXB_GMIN 3.33e-07f
#define CRXB_DG   (CRXB_GMAX - CRXB_GMIN)
#define CRXB_DV   (3.3f / 255.0f)

// out[n, col*64+s, r] = sum_k g[col,r,s,k] * x[n,r,k] * DELTA_V
// x: [4096, 16, 64]  w: [16, 16, 64, 64]  out: [4096, 1024, 16]
// One workgroup = 512 threads = 16 waves; wave w handles crossbar row r = w
// for one (col, 16-batch tile). Results staged in 64KB LDS, stored coalesced.
__global__ __launch_bounds__(512) void crxb_wmma_kernel(
    const float* __restrict__ x, const float* __restrict__ w,
    float* __restrict__ out)
{
  extern __shared__ float lds[];          // 16 * 1024 floats = 64 KB
  const int lane = threadIdx.x & 31;
  const int r    = threadIdx.x >> 5;      // wave id == crossbar row, 0..15
  const int jn   = lane & 15;             // M / N index within fragment
  const int hi   = lane >> 4;             // K-half select (per ISA A/B layout)
  const int col  = blockIdx.y;
  const int n0   = blockIdx.x * 16;

  // ---- B fragments: input tile (shared across all 4 m-tiles) ----
  // lane jn -> batch n0+jn; VGPR pair holds K = 4*kk + 2*hi + {0,1}
  const float* xb = x + (long)(n0 + jn) * 1024 + r * 64 + hi * 2;
  v2f bf[16];
#pragma unroll
  for (int kk = 0; kk < 16; ++kk) {
    v2f t = *(const v2f*)(xb + kk * 4);   // global_load_b64 per lane
    bf[kk] = t * CRXB_DV;                 // DAC scaling
  }

  // weight base for (col, r): lane jn -> s = m*16 + jn, same K packing as B
  const float* wb = w + ((long)(col * 16 + r) * 64 + jn) * 64 + hi * 2;

  for (int m = 0; m < 4; ++m) {
    // ---- A fragments: signed weight -> pos/neg conductance -> difference ----
    v2f af[16];
    const float* wp = wb + m * 16 * 64;
#pragma unroll
    for (int kk = 0; kk < 16; ++kk) {
      v2f t = *(const v2f*)(wp + kk * 4);
      float gpx = CRXB_GMIN + fmaxf(t.x, 0.0f) * CRXB_DG;
      float gpy = CRXB_GMIN + fmaxf(t.y, 0.0f) * CRXB_DG;
      float gnx = CRXB_GMIN + fmaxf(-t.x, 0.0f) * CRXB_DG;
      float gny = CRXB_GMIN + fmaxf(-t.y, 0.0f) * CRXB_DG;
      v2f g; g.x = gpx - gnx; g.y = gpy - gny;
      af[kk] = g;
    }

    // ---- 16 chained fp32 WMMAs: D(16s x 16n) += A(16x4) * B(4x16) ----
    v8f acc = {0.f, 0.f, 0.f, 0.f, 0.f, 0.f, 0.f, 0.f};
#pragma unroll
    for (int kk = 0; kk < 16; ++kk) {
      acc = __builtin_amdgcn_wmma_f32_16x16x4_f32(
          false, af[kk], false, bf[kk], (short)0, acc, false, false);
    }

    // ---- D -> LDS (swizzled: +34 floats per batch row, keeps b64 pairs) ----
    // D layout: VGPR v, lanes 0-15: M=v, N=lane; lanes 16-31: M=v+8, N=lane-16
    const int nn = jn;                    // batch within tile
#pragma unroll
    for (int v = 0; v < 8; ++v) {
      int s = m * 16 + v + hi * 8;
      int q = s * 16 + r;                 // position within the 1024-float block
      lds[nn * 1024 + ((q + nn * 34) & 1023)] = acc[v];
    }
  }
  __syncthreads();

  // ---- coalesced flush: per batch element, 1024 contiguous floats ----
  float* ob = out + (long)n0 * 16384 + col * 1024;
  const int t = threadIdx.x;
  const int q = 2 * t;
#pragma unroll 4
  for (int nn = 0; nn < 16; ++nn) {
    v2f d = *(const v2f*)&lds[nn * 1024 + ((q + nn * 34) & 1023)];
    *(v2f*)(ob + (long)nn * 16384 + q) = d;   // global_store_b64, 4KB/iter/WG
  }
}

extern "C" void kernel_launch(void* const* d_in, const int* in_sizes, int n_in,
                              void* d_out, int out_size, void* d_ws, size_t ws_size,
                              hipStream_t stream) {
  (void)in_sizes; (void)n_in; (void)d_ws; (void)ws_size; (void)out_size;
  const float* x = (const float*)d_in[0];   // [4096,1,16,64,1] f32
  const float* w = (const float*)d_in[1];   // [16,16,64,64]    f32
  // d_in[2] (nchout_index) is arange -> identity gather, ignored.
  float* out = (float*)d_out;               // [4096,1024,16]   f32

  dim3 grid(4096 / 16, 16);                 // 256 batch tiles x 16 cols
  dim3 block(512);                          // 16 waves (wave32)
  hipLaunchKernelGGL(crxb_wmma_kernel, grid, block,
                     16 * 1024 * sizeof(float), stream, x, w, out);
}